// WordAttention_43508018708772
// MI455X (gfx1250) — compile-verified
//
#include <hip/hip_runtime.h>
#include <math.h>

#define Bq 256
#define Tq 512
#define Dq 512

typedef __attribute__((ext_vector_type(2))) float v2f;
typedef __attribute__((ext_vector_type(8))) float v8f;

// ---------------------------------------------------------------------------
// Kernel A: inter[B,D] = hidden[B,D] @ W[D,D] using V_WMMA_F32_16X16X4_F32.
// One wave (32 threads) computes one 16x16 tile of inter; K-loop 512/4 = 128
// WMMA accumulations. Tiny kernel (67 MFLOP), runs on the matrix pipe.
// ---------------------------------------------------------------------------
__global__ __launch_bounds__(32) void inter_gemm_wmma(
    const float* __restrict__ hidden, const float* __restrict__ W,
    float* __restrict__ inter)
{
    const int lane = threadIdx.x;   // 0..31 (wave32)
    const int lo   = lane & 15;
    const int hi   = lane >> 4;     // 0 or 1
    const int row0 = blockIdx.y << 4;   // batch-row tile
    const int col0 = blockIdx.x << 4;   // output-col tile

    v8f acc = {};
    const float* arow = hidden + (size_t)(row0 + lo) * Dq;

    for (int k = 0; k < Dq; k += 4) {
        // A (16x4 f32): VGPR j holds K = 2*hi + j for row M = lo  -> float2 load
        v2f a = *(const v2f*)(arow + k + 2 * hi);
        // B (4x16 f32): VGPR j holds row K = 2*hi + j, col N = lo
        v2f b;
        b.x = W[(size_t)(k + 2 * hi)     * Dq + col0 + lo];
        b.y = W[(size_t)(k + 2 * hi + 1) * Dq + col0 + lo];
        acc = __builtin_amdgcn_wmma_f32_16x16x4_f32(
            /*neg_a=*/false, a, /*neg_b=*/false, b,
            /*c_mod=*/(short)0, acc, /*reuse_a=*/false, /*reuse_b=*/false);
    }

    // C/D layout: element (M = v + 8*hi, N = lo) lives in acc[v]
    #pragma unroll
    for (int v = 0; v < 8; ++v) {
        inter[(size_t)(row0 + v + 8 * hi) * Dq + col0 + lo] = acc[v];
    }
}

// ---------------------------------------------------------------------------
// Kernel B: per-batch GEMV + softmax + alpha scale. Bandwidth-bound: streams
// 256 MB of `words` once with fully coalesced float4 (global_load_b128) reads.
// One block (512 threads = 16 waves) per batch. Each wave computes 32 rows'
// dot products. Each lane's 16-float slice of inter[b,:] is held in
// registers (loop-invariant) -- no LDS traffic in the streaming loop.
// ---------------------------------------------------------------------------
__global__ __launch_bounds__(512) void scores_softmax(
    const float* __restrict__ words, const float* __restrict__ inter,
    const float* __restrict__ alpha, float* __restrict__ out)
{
    __shared__ float s_scores[Tq];
    __shared__ float s_red[512];

    const int tid  = threadIdx.x;
    const int b    = blockIdx.x;
    const int wave = tid >> 5;
    const int lane = tid & 31;

    // Hoist this lane's slice of inter[b,:] into registers (coalesced loads,
    // identical across the 16 waves -> L0/L2 hits).
    const float4* si4 = (const float4*)(inter + (size_t)b * Dq);
    float4 iv[4];
    #pragma unroll
    for (int i = 0; i < 4; ++i) iv[i] = si4[lane + 32 * i];

    #pragma unroll 4
    for (int r = 0; r < 32; ++r) {
        const int t = wave * 32 + r;
        const float4* wrow = (const float4*)(words + ((size_t)b * Tq + t) * Dq);

        // Prefetch this wave's next row (global_prefetch_b8 path)
        if (r < 31) {
            const char* np = (const char*)(words + ((size_t)b * Tq + t + 1) * Dq);
            __builtin_prefetch(np + lane * 64, 0, 1);
        }

        float acc = 0.f;
        #pragma unroll
        for (int i = 0; i < 4; ++i) {
            float4 wv = wrow[lane + 32 * i];   // 512B per wave per load: coalesced
            acc += wv.x * iv[i].x + wv.y * iv[i].y + wv.z * iv[i].z + wv.w * iv[i].w;
        }
        // wave32 reduction
        #pragma unroll
        for (int m = 16; m >= 1; m >>= 1)
            acc += __shfl_xor(acc, m, 32);
        if (lane == 0) s_scores[t] = acc;
    }
    __syncthreads();

    // Softmax over T = 512 (one score per thread)
    const float x = s_scores[tid];

    s_red[tid] = x;
    __syncthreads();
    for (int s = 256; s >= 1; s >>= 1) {
        if (tid < s) s_red[tid] = fmaxf(s_red[tid], s_red[tid + s]);
        __syncthreads();
    }
    const float mx = s_red[0];
    __syncthreads();

    const float e = __expf(x - mx);
    s_red[tid] = e;
    __syncthreads();
    for (int s = 256; s >= 1; s >>= 1) {
        if (tid < s) s_red[tid] += s_red[tid + s];
        __syncthreads();
    }
    const float scale = alpha[b] / s_red[0];
    out[(size_t)b * Tq + tid] = e * scale;
}

extern "C" void kernel_launch(void* const* d_in, const int* in_sizes, int n_in,
                              void* d_out, int out_size, void* d_ws, size_t ws_size,
                              hipStream_t stream) {
    // setup_inputs order: hidden [B,D], words [B,T,D], alpha [B,1], W_words [D,D], batch_size
    const float* hidden = (const float*)d_in[0];
    const float* words  = (const float*)d_in[1];
    const float* alpha  = (const float*)d_in[2];
    const float* W      = (const float*)d_in[3];
    float* out   = (float*)d_out;
    float* inter = (float*)d_ws;   // 256*512*4 = 512 KB scratch

    dim3 gridA(Dq / 16, Bq / 16);          // 32 x 16 = 512 tiles, 1 wave each
    inter_gemm_wmma<<<gridA, 32, 0, stream>>>(hidden, W, inter);

    scores_softmax<<<Bq, 512, 0, stream>>>(words, inter, alpha, out);
}